// Net_59768764891998
// MI455X (gfx1250) — compile-verified
//
#include <hip/hip_runtime.h>
#include <hip/hip_bf16.h>
#include <math.h>

// SplineConv 2-layer GNN forward for MI455X (gfx1250, wave32).
// Layer-1 GEMM uses V_WMMA_F32_16X16X4_F32 on the matrix pipe with a
// pre-packed (fragment-ordered) B so every B fetch is one coalesced b64.

typedef float v2f __attribute__((ext_vector_type(2)));
typedef float v8f __attribute__((ext_vector_type(8)));

#define F_IN 767
#define HID  16
#define CLS  10
#define KSTEPS 192          // ceil(767/4)
#define BP_FLOATS (KSTEPS * 3 * 64)   // 36864 floats = 147KB

#define WMMA_F32(A, B, C) \
  __builtin_amdgcn_wmma_f32_16x16x4_f32(false, (A), false, (B), (short)0, (C), false, false)

// ---------------------------------------------------------------------------
// Prep: pack [ W1[0] | W1[1] | root1 ] (each 767x16) into WMMA B-fragment
// order with zero padding at k=767.
// B frag (4x16 f32): lane l -> col (l&15), VGPR j holds K = (l>>4)*2 + j.
// Bp[((ks*3 + tile)*32 + lane)*2 + j]
// ---------------------------------------------------------------------------
__global__ void pack_W1(const float* __restrict__ W1, const float* __restrict__ root1,
                        float* __restrict__ Bp)
{
  int t = blockIdx.x * blockDim.x + threadIdx.x;
  if (t >= KSTEPS * 3 * 32) return;
  const int lane = t & 31;
  const int tile = (t >> 5) % 3;
  const int ks   = t / 96;
  const int n    = lane & 15;
  const int k0   = ks * 4 + ((lane >> 4) << 1);
  const float* B = (tile == 0) ? W1 : (tile == 1) ? (W1 + F_IN * HID) : root1;
  const float v0 = (k0     < F_IN) ? B[k0 * HID + n]       : 0.0f;
  const float v1 = (k0 + 1 < F_IN) ? B[(k0 + 1) * HID + n] : 0.0f;
  const long  o  = ((long)ks * 3 + tile) * 64 + lane * 2;
  Bp[o]     = v0;
  Bp[o + 1] = v1;
}

// ---------------------------------------------------------------------------
// Kernel 1: H[N,48] = x[N,767] @ [ W1[0] | W1[1] | root1 ]
// One wave per 32-row job (2 M-tiles); 6 accumulators; B fragments shared
// across both M-tiles. Per k-step: 2x A(b64) + 3x B(b64) feed 6 WMMAs.
// A frag (16x4 f32): lane l -> row m+(l&15), VGPR j holds K = (l>>4)*2 + j.
// C/D (16x16 f32):   VGPR r -> row m + r + (l>>4)*8, col (l&15).
// ---------------------------------------------------------------------------
__global__ __launch_bounds__(128)
void gemm1_wmma(const float* __restrict__ x, const float* __restrict__ Bp,
                float* __restrict__ H, int N, int jobs)
{
  const int lane = threadIdx.x & 31;
  const int wave = threadIdx.x >> 5;
  const int job  = blockIdx.x * 4 + wave;
  if (job >= jobs) return;                 // wave-uniform exit (EXEC all-1 for WMMA)

  const int m0    = job * 32;
  const int khalf = (lane >> 4) << 1;      // 0 or 2
  const int ncol  = lane & 15;
  int r0 = m0 + ncol;        if (r0 >= N) r0 = N - 1;   // clamp keeps lanes active
  int r1 = m0 + 16 + ncol;   if (r1 >= N) r1 = N - 1;

  const float* __restrict__ a0p = x + (long)r0 * F_IN;
  const float* __restrict__ a1p = x + (long)r1 * F_IN;
  const float* __restrict__ bpl = Bp + lane * 2;

  v8f c00 = {}; v8f c01 = {}; v8f c02 = {};
  v8f c10 = {}; v8f c11 = {}; v8f c12 = {};

  int ks = 0;
  #pragma unroll 2
  for (; ks < KSTEPS - 1; ++ks) {          // 191 full steps (k = 0..763)
    const int  kb = ks * 4 + khalf;
    const long bo = (long)ks * 192;
    if ((ks & 7) == 0) {
      __builtin_prefetch(a0p + kb + 256, 0, 0);   // global_prefetch_b8
      __builtin_prefetch(a1p + kb + 256, 0, 0);
    }
    v2f a0; a0.x = a0p[kb];        a0.y = a0p[kb + 1];
    v2f a1; a1.x = a1p[kb];        a1.y = a1p[kb + 1];
    v2f b0; b0.x = bpl[bo];        b0.y = bpl[bo + 1];
    v2f b1; b1.x = bpl[bo + 64];   b1.y = bpl[bo + 65];
    v2f b2; b2.x = bpl[bo + 128];  b2.y = bpl[bo + 129];
    c00 = WMMA_F32(a0, b0, c00);
    c10 = WMMA_F32(a1, b0, c10);
    c01 = WMMA_F32(a0, b1, c01);
    c11 = WMMA_F32(a1, b1, c11);
    c02 = WMMA_F32(a0, b2, c02);
    c12 = WMMA_F32(a1, b2, c12);
  }
  { // peeled step ks=191: k = 764..767; k=767 column of Bp is zero, so the
    // clamped A element contributes exactly 0 (finite * 0).
    const int  kb = (KSTEPS - 1) * 4 + khalf;            // 764 or 766
    const int  k1 = (kb + 1 < F_IN) ? kb + 1 : F_IN - 1; // 765 or clamp(767)->766
    const long bo = (long)(KSTEPS - 1) * 192;
    v2f a0; a0.x = a0p[kb];        a0.y = a0p[k1];
    v2f a1; a1.x = a1p[kb];        a1.y = a1p[k1];
    v2f b0; b0.x = bpl[bo];        b0.y = bpl[bo + 1];
    v2f b1; b1.x = bpl[bo + 64];   b1.y = bpl[bo + 65];
    v2f b2; b2.x = bpl[bo + 128];  b2.y = bpl[bo + 129];
    c00 = WMMA_F32(a0, b0, c00);
    c10 = WMMA_F32(a1, b0, c10);
    c01 = WMMA_F32(a0, b1, c01);
    c11 = WMMA_F32(a1, b1, c11);
    c02 = WMMA_F32(a0, b2, c02);
    c12 = WMMA_F32(a1, b2, c12);
  }

  const int rbase = (lane >> 4) << 3;
  #pragma unroll
  for (int r = 0; r < 8; ++r) {
    const int orow0 = m0 + rbase + r;
    if (orow0 < N) {
      const long base = (long)orow0 * 48;
      H[base +      ncol] = c00[r];   // h0 = x @ W1[0]
      H[base + 16 + ncol] = c01[r];   // h1 = x @ W1[1]
      H[base + 32 + ncol] = c02[r];   // hr = x @ root1
    }
    const int orow1 = orow0 + 16;
    if (orow1 < N) {
      const long base = (long)orow1 * 48;
      H[base +      ncol] = c10[r];
      H[base + 16 + ncol] = c11[r];
      H[base + 32 + ncol] = c12[r];
    }
  }
}

// ---------------------------------------------------------------------------
__global__ void zero_f32(float* __restrict__ p, long n)
{
  long i = (long)blockIdx.x * blockDim.x + threadIdx.x;
  if (i < n) p[i] = 0.0f;
}

// Layer-1 scatter: lane = (edge, feature). 16 features per edge.
__global__ void edge_scatter1(const int* __restrict__ ei, const float* __restrict__ u,
                              const float* __restrict__ H, float* __restrict__ agg,
                              float* __restrict__ deg, int E)
{
  long t = (long)blockIdx.x * blockDim.x + threadIdx.x;
  long e = t >> 4;
  int  n = (int)(t & 15);
  if (e >= E) return;
  const int   src = ei[e];
  const int   dst = ei[(long)E + e];
  const float uu  = u[e];
  const long  hb  = (long)src * 48;
  const float m   = (1.0f - uu) * H[hb + n] + uu * H[hb + 16 + n];
  atomicAdd(&agg[(long)dst * HID + n], m);
  if (n == 0) atomicAdd(&deg[dst], 1.0f);
}

// Finalize layer 1 (x1 = relu(agg/deg + x@root1 + b1)) fused with the tiny
// layer-2 GEMM: G[i,0:30] = x1[i] @ [ W2[0] | W2[1] | root2 ].
__global__ void finalize1(const float* __restrict__ agg, const float* __restrict__ deg,
                          const float* __restrict__ H, const float* __restrict__ b1,
                          const float* __restrict__ W2, const float* __restrict__ root2,
                          float* __restrict__ x1_out, float* __restrict__ G, int N)
{
  int i = blockIdx.x * blockDim.x + threadIdx.x;
  if (i >= N) return;
  float d   = deg[i];
  float inv = 1.0f / (d > 1.0f ? d : 1.0f);
  float x1[HID];
  #pragma unroll
  for (int n = 0; n < HID; ++n) {
    float v = agg[(long)i * HID + n] * inv + H[(long)i * 48 + 32 + n] + b1[n];
    x1[n] = v > 0.0f ? v : 0.0f;
    x1_out[(long)i * HID + n] = x1[n];
  }
  #pragma unroll
  for (int c = 0; c < CLS; ++c) {
    float s0 = 0.0f, s1 = 0.0f, sr = 0.0f;
    #pragma unroll
    for (int k = 0; k < HID; ++k) {
      s0 += x1[k] * W2[k * CLS + c];              // W2[0]
      s1 += x1[k] * W2[HID * CLS + k * CLS + c];  // W2[1]
      sr += x1[k] * root2[k * CLS + c];
    }
    G[(long)i * 30 +      c] = s0;
    G[(long)i * 30 + 10 + c] = s1;
    G[(long)i * 30 + 20 + c] = sr;
  }
}

// Layer-2 scatter: 10 features per edge (6/16 lanes idle — acceptable).
__global__ void edge_scatter2(const int* __restrict__ ei, const float* __restrict__ u,
                              const float* __restrict__ G, float* __restrict__ agg2, int E)
{
  long t = (long)blockIdx.x * blockDim.x + threadIdx.x;
  long e = t >> 4;
  int  n = (int)(t & 15);
  if (e >= E) return;
  if (n >= CLS) return;
  const int   src = ei[e];
  const int   dst = ei[(long)E + e];
  const float uu  = u[e];
  const long  gb  = (long)src * 30;
  const float m   = (1.0f - uu) * G[gb + n] + uu * G[gb + 10 + n];
  atomicAdd(&agg2[(long)dst * CLS + n], m);
}

// Finalize layer 2 + log_softmax.
__global__ void finalize2(const float* __restrict__ agg2, const float* __restrict__ deg,
                          const float* __restrict__ G, const float* __restrict__ b2,
                          float* __restrict__ out, int N)
{
  int i = blockIdx.x * blockDim.x + threadIdx.x;
  if (i >= N) return;
  float d   = deg[i];
  float inv = 1.0f / (d > 1.0f ? d : 1.0f);
  float v[CLS];
  float mx = -3.4e38f;
  #pragma unroll
  for (int c = 0; c < CLS; ++c) {
    v[c] = agg2[(long)i * CLS + c] * inv + G[(long)i * 30 + 20 + c] + b2[c];
    mx = v[c] > mx ? v[c] : mx;
  }
  float s = 0.0f;
  #pragma unroll
  for (int c = 0; c < CLS; ++c) s += expf(v[c] - mx);
  const float ls = logf(s);
  #pragma unroll
  for (int c = 0; c < CLS; ++c) out[(long)i * CLS + c] = v[c] - mx - ls;
}

// ---------------------------------------------------------------------------
extern "C" void kernel_launch(void* const* d_in, const int* in_sizes, int n_in,
                              void* d_out, int out_size, void* d_ws, size_t ws_size,
                              hipStream_t stream)
{
  const float* x     = (const float*)d_in[0];
  const int*   ei    = (const int*)  d_in[1];   // harness passes integers as int32
  const float* u     = (const float*)d_in[2];
  const float* W1    = (const float*)d_in[3];   // [2,767,16]
  const float* root1 = (const float*)d_in[4];   // [767,16]
  const float* b1    = (const float*)d_in[5];
  const float* W2    = (const float*)d_in[6];   // [2,16,10]
  const float* root2 = (const float*)d_in[7];   // [16,10]
  const float* b2    = (const float*)d_in[8];

  const int N = in_sizes[0] / F_IN;
  const int E = in_sizes[1] / 2;

  // Workspace layout (floats): H | agg1 | deg | agg2 | G | Bp   (~42.2 MB)
  float* ws   = (float*)d_ws;
  float* H    = ws;                          // N*48
  float* agg1 = H    + (long)N * 48;         // N*16
  float* deg  = agg1 + (long)N * HID;        // N      (contiguous with agg1/agg2)
  float* agg2 = deg  + (long)N;              // N*10
  float* G    = agg2 + (long)N * CLS;        // N*30
  float* Bp   = G    + (long)N * 30;         // 36864 (packed WMMA B fragments)

  float* out_ls = (float*)d_out;             // log_softmax : N*10  (output 0)
  float* out_x1 = out_ls + (long)N * CLS;    // x1          : N*16  (output 1)

  pack_W1<<<(KSTEPS * 3 * 32 + 255) / 256, 256, 0, stream>>>(W1, root1, Bp);

  const int mtiles = (N + 15) / 16;
  const int jobs   = (mtiles + 1) / 2;       // 32 rows per wave
  gemm1_wmma<<<(jobs + 3) / 4, 128, 0, stream>>>(x, Bp, H, N, jobs);

  const long nz = (long)N * (HID + 1 + CLS); // zero agg1|deg|agg2 every call
  zero_f32<<<(int)((nz + 255) / 256), 256, 0, stream>>>(agg1, nz);

  const long t1 = (long)E * 16;
  edge_scatter1<<<(int)((t1 + 255) / 256), 256, 0, stream>>>(ei, u, H, agg1, deg, E);

  finalize1<<<(N + 255) / 256, 256, 0, stream>>>(agg1, deg, H, b1, W2, root2, out_x1, G, N);

  edge_scatter2<<<(int)((t1 + 255) / 256), 256, 0, stream>>>(ei, u, G, agg2, E);

  finalize2<<<(N + 255) / 256, 256, 0, stream>>>(agg2, deg, G, b2, out_ls, N);
}